// WalshSpectrumEncoder_24266565222503
// MI455X (gfx1250) — compile-verified
//
#include <hip/hip_runtime.h>

typedef float v2f __attribute__((ext_vector_type(2)));
typedef float v8f __attribute__((ext_vector_type(8)));

namespace {
constexpr int kBatch = 524288;
constexpr int kThreads = 128;        // 4 waves (wave32)
constexpr int kRowsPerBlock = 128;   // 1 row per thread in feature stage
constexpr int kF = 36;               // 33 features padded to 36 (K multiple of 4)

struct __align__(16) Smem {
    float w1t[64 * kF];      // W1 transposed [n][k], k 33..35 zero   (9216 B)
    float w2t[64 * 64];      // W2 transposed [n][k]                  (16384 B)
    float w3t[64 * 64];      // W3 transposed [n][k]                  (16384 B)
    float feats[kRowsPerBlock * kF];                               // (18432 B)
    float act[4 * 16 * 68];  // per-wave 16x64 activations, stride 68 (17408 B)
};
} // namespace

__device__ __forceinline__ v8f wmma_f32_4(v2f a, v2f b, v8f c) {
    // D(16x16,f32) = A(16x4,f32) * B(4x16,f32) + C
    return __builtin_amdgcn_wmma_f32_16x16x4_f32(false, a, false, b, (short)0, c,
                                                 false, false);
}

__device__ __forceinline__ float gelu_exact(float x) {
    return 0.5f * x * (1.0f + erff(x * 0.70710678118654752440f));
}

// LayerNorm + exact GELU on a 16x64 D-tile held as 4 accumulators, then store
// into per-wave LDS in row-major [16][68] (A-matrix friendly) layout.
// D layout: vgpr v, lanes 0-15 -> M=v,N=lane; lanes 16-31 -> M=v+8,N=lane-16.
__device__ __forceinline__ void ln_gelu_store(v8f c0, v8f c1, v8f c2, v8f c3,
                                              const float* vb, const float* vg,
                                              const float* vbe, float* act,
                                              int l15, int lh) {
#pragma unroll
    for (int v = 0; v < 8; ++v) {
        float x0 = c0[v] + vb[0];
        float x1 = c1[v] + vb[1];
        float x2 = c2[v] + vb[2];
        float x3 = c3[v] + vb[3];
        // mean over N=64: sum 4 n-tiles then butterfly over the 16-lane half
        float s = x0 + x1 + x2 + x3;
#pragma unroll
        for (int m = 1; m <= 8; m <<= 1) s += __shfl_xor(s, m, 32);
        float mean = s * (1.0f / 64.0f);
        float t0 = x0 - mean, t1 = x1 - mean, t2 = x2 - mean, t3 = x3 - mean;
        float q = t0 * t0 + t1 * t1 + t2 * t2 + t3 * t3;
#pragma unroll
        for (int m = 1; m <= 8; m <<= 1) q += __shfl_xor(q, m, 32);
        float rstd = rsqrtf(q * (1.0f / 64.0f) + 1e-5f);
        int ro = (v + 8 * lh) * 68;
        act[ro + 0 * 16 + l15] = gelu_exact(t0 * rstd * vg[0] + vbe[0]);
        act[ro + 1 * 16 + l15] = gelu_exact(t1 * rstd * vg[1] + vbe[1]);
        act[ro + 2 * 16 + l15] = gelu_exact(t2 * rstd * vg[2] + vbe[2]);
        act[ro + 3 * 16 + l15] = gelu_exact(t3 * rstd * vg[3] + vbe[3]);
    }
}

extern "C" __global__ __launch_bounds__(kThreads)
void walsh_encoder_kernel(const int* __restrict__ tt,
                          const float* __restrict__ W1, const float* __restrict__ b1,
                          const float* __restrict__ g1, const float* __restrict__ be1,
                          const float* __restrict__ W2, const float* __restrict__ b2,
                          const float* __restrict__ g2, const float* __restrict__ be2,
                          const float* __restrict__ W3, const float* __restrict__ b3,
                          float* __restrict__ out) {
    __shared__ Smem sm;
    const int tid = threadIdx.x;
    const int lane = tid & 31;
    const int wid = tid >> 5;
    const int l15 = lane & 15;
    const int lh = lane >> 4;

    // ---------------- stage weights (transposed) into LDS ----------------
    for (int idx = tid; idx < 64 * kF; idx += kThreads) {
        int n = idx / kF, k = idx - n * kF;
        sm.w1t[idx] = (k < 33) ? W1[k * 64 + n] : 0.0f;
    }
    for (int idx = tid; idx < 64 * 64; idx += kThreads) {
        int n = idx >> 6, k = idx & 63;
        sm.w2t[idx] = W2[k * 64 + n];
        sm.w3t[idx] = W3[k * 64 + n];
    }

    // ---------------- per-thread Walsh feature stage ----------------
    {
        const long long row = (long long)blockIdx.x * kRowsPerBlock + tid;
        const int4* p4 = reinterpret_cast<const int4*>(tt + row * 64);
        float w[64];
#pragma unroll
        for (int i = 0; i < 16; ++i) {
            int4 v = p4[i];
            w[4 * i + 0] = 1.0f - 2.0f * (float)v.x;
            w[4 * i + 1] = 1.0f - 2.0f * (float)v.y;
            w[4 * i + 2] = 1.0f - 2.0f * (float)v.z;
            w[4 * i + 3] = 1.0f - 2.0f * (float)v.w;
        }
        // fast Walsh-Hadamard transform, fully unrolled (H symmetric)
#pragma unroll
        for (int s = 1; s < 64; s <<= 1) {
#pragma unroll
            for (int i = 0; i < 64; ++i) {
                if ((i & s) == 0) {
                    float a = w[i], b = w[i | s];
                    w[i] = a + b;
                    w[i | s] = a - b;
                }
            }
        }
        float* fr = &sm.feats[tid * kF];
        const float inv_n = 1.0f / 64.0f;
        fr[0] = w[0] * inv_n;  fr[1] = w[1] * inv_n;
        fr[2] = w[2] * inv_n;  fr[3] = w[4] * inv_n;
        fr[4] = w[8] * inv_n;  fr[5] = w[16] * inv_n;
        fr[6] = w[32] * inv_n; fr[7] = w[63] * inv_n;
        // abs in place (signed values no longer needed)
#pragma unroll
        for (int i = 0; i < 64; ++i) w[i] = fabsf(w[i]);

        float cnt_small = 0.0f, mx = 0.0f, l1 = 0.0f, ssq = 0.0f;
        float cnt_large = 0.0f, degw = 0.0f, degc = 0.0f, bdev = 0.0f;
#pragma unroll
        for (int i = 0; i < 64; ++i) {
            float a = w[i];
            cnt_small += (a < 6.4f) ? 1.0f : 0.0f;
            mx = fmaxf(mx, a);
            l1 += a;
            ssq += a * a;
            cnt_large += (a > 32.0f) ? 1.0f : 0.0f;
            float lm = (a > 6.4f) ? 1.0f : 0.0f;
            degw += lm * (float)__builtin_popcount(i);
            degc += lm;
            bdev += fabsf(a - 8.0f);
        }
        float inv_ssq = 1.0f / (ssq + 1e-10f);
        float ent = 0.0f;
#pragma unroll
        for (int i = 0; i < 64; ++i) {
            float p = w[i] * w[i] * inv_ssq;
            ent += p * __logf(p + 1e-10f);
        }
        fr[8] = cnt_small * (1.0f / 64.0f);
        fr[9] = mx * inv_n;
        fr[10] = l1 * (1.0f / 4096.0f);
        fr[11] = sqrtf(ssq) * (1.0f / 4096.0f);
        fr[12] = -ent * (1.0f / 4.1588830833596715f);  // / log(64)
        fr[13] = (cnt_large >= 1.0f && cnt_large <= 2.0f) ? 1.0f : 0.0f;
        float nl = fminf(fmaxf(0.5f - mx * (1.0f / 128.0f), 0.0f), 0.5f);
        fr[14] = nl * 2.0f;
        fr[15] = degw / (degc + 1e-10f) * (1.0f / 6.0f);
        fr[16] = ((bdev * (1.0f / 512.0f)) < 0.1f) ? 1.0f : 0.0f;

        // bitonic sort |W| descending, static comparator network
#pragma unroll
        for (int k = 2; k <= 64; k <<= 1) {
#pragma unroll
            for (int j = k >> 1; j > 0; j >>= 1) {
#pragma unroll
                for (int i = 0; i < 64; ++i) {
                    int l = i ^ j;
                    if (l > i) {
                        float a = w[i], b = w[l];
                        float mn = fminf(a, b), mxv = fmaxf(a, b);
                        if ((i & k) == 0) { w[i] = mxv; w[l] = mn; }
                        else              { w[i] = mn;  w[l] = mxv; }
                    }
                }
            }
        }
#pragma unroll
        for (int i = 0; i < 16; ++i) fr[17 + i] = w[i] * inv_n;
        fr[33] = 0.0f; fr[34] = 0.0f; fr[35] = 0.0f;
    }
    __syncthreads();

    // ---------------- per-wave WMMA MLP stage ----------------
    float vb1[4], vg1[4], vbe1[4], vb2[4], vg2[4], vbe2[4], vb3[4];
#pragma unroll
    for (int nt = 0; nt < 4; ++nt) {
        int n = nt * 16 + l15;
        vb1[nt] = b1[n]; vg1[nt] = g1[n]; vbe1[nt] = be1[n];
        vb2[nt] = b2[n]; vg2[nt] = g2[n]; vbe2[nt] = be2[n];
        vb3[nt] = b3[n];
    }
    float* act = &sm.act[wid * 16 * 68];

    for (int tile = 0; tile < 2; ++tile) {
        const int rowBase = wid * 32 + tile * 16;  // block-local first row

        // ---- layer 1: feats[16 x 36] @ W1 -> 16 x 64 ----
        v8f c0 = {}, c1 = {}, c2 = {}, c3 = {};
#pragma unroll
        for (int k0 = 0; k0 < kF; k0 += 4) {
            // A frag: lane -> M=l15, K = k0 + 2*lh (+1)
            v2f a = *(const v2f*)&sm.feats[(rowBase + l15) * kF + k0 + 2 * lh];
            // B frag: lane -> N=l15 (+16*nt), K = k0 + 2*lh (+1) from transposed W
            v2f bw0 = *(const v2f*)&sm.w1t[(0 * 16 + l15) * kF + k0 + 2 * lh];
            v2f bw1 = *(const v2f*)&sm.w1t[(1 * 16 + l15) * kF + k0 + 2 * lh];
            v2f bw2 = *(const v2f*)&sm.w1t[(2 * 16 + l15) * kF + k0 + 2 * lh];
            v2f bw3 = *(const v2f*)&sm.w1t[(3 * 16 + l15) * kF + k0 + 2 * lh];
            c0 = wmma_f32_4(a, bw0, c0);
            c1 = wmma_f32_4(a, bw1, c1);
            c2 = wmma_f32_4(a, bw2, c2);
            c3 = wmma_f32_4(a, bw3, c3);
        }
        ln_gelu_store(c0, c1, c2, c3, vb1, vg1, vbe1, act, l15, lh);

        // ---- layer 2: act[16 x 64] @ W2 -> 16 x 64 ----
        c0 = {}; c1 = {}; c2 = {}; c3 = {};
#pragma unroll
        for (int k0 = 0; k0 < 64; k0 += 4) {
            v2f a = *(const v2f*)&act[l15 * 68 + k0 + 2 * lh];
            v2f bw0 = *(const v2f*)&sm.w2t[(0 * 16 + l15) * 64 + k0 + 2 * lh];
            v2f bw1 = *(const v2f*)&sm.w2t[(1 * 16 + l15) * 64 + k0 + 2 * lh];
            v2f bw2 = *(const v2f*)&sm.w2t[(2 * 16 + l15) * 64 + k0 + 2 * lh];
            v2f bw3 = *(const v2f*)&sm.w2t[(3 * 16 + l15) * 64 + k0 + 2 * lh];
            c0 = wmma_f32_4(a, bw0, c0);
            c1 = wmma_f32_4(a, bw1, c1);
            c2 = wmma_f32_4(a, bw2, c2);
            c3 = wmma_f32_4(a, bw3, c3);
        }
        ln_gelu_store(c0, c1, c2, c3, vb2, vg2, vbe2, act, l15, lh);

        // ---- layer 3: act[16 x 64] @ W3 + b3 -> out ----
        c0 = {}; c1 = {}; c2 = {}; c3 = {};
#pragma unroll
        for (int k0 = 0; k0 < 64; k0 += 4) {
            v2f a = *(const v2f*)&act[l15 * 68 + k0 + 2 * lh];
            v2f bw0 = *(const v2f*)&sm.w3t[(0 * 16 + l15) * 64 + k0 + 2 * lh];
            v2f bw1 = *(const v2f*)&sm.w3t[(1 * 16 + l15) * 64 + k0 + 2 * lh];
            v2f bw2 = *(const v2f*)&sm.w3t[(2 * 16 + l15) * 64 + k0 + 2 * lh];
            v2f bw3 = *(const v2f*)&sm.w3t[(3 * 16 + l15) * 64 + k0 + 2 * lh];
            c0 = wmma_f32_4(a, bw0, c0);
            c1 = wmma_f32_4(a, bw1, c1);
            c2 = wmma_f32_4(a, bw2, c2);
            c3 = wmma_f32_4(a, bw3, c3);
        }
#pragma unroll
        for (int v = 0; v < 8; ++v) {
            long long gr = (long long)blockIdx.x * kRowsPerBlock + rowBase + v + 8 * lh;
            float* o = out + gr * 64;
            o[0 * 16 + l15] = c0[v] + vb3[0];
            o[1 * 16 + l15] = c1[v] + vb3[1];
            o[2 * 16 + l15] = c2[v] + vb3[2];
            o[3 * 16 + l15] = c3[v] + vb3[3];
        }
    }
}

extern "C" void kernel_launch(void* const* d_in, const int* in_sizes, int n_in,
                              void* d_out, int out_size, void* d_ws, size_t ws_size,
                              hipStream_t stream) {
    const int* tt = (const int*)d_in[0];
    // d_in[1] = H (unused: FWHT computed directly)
    const float* W1 = (const float*)d_in[2];
    const float* b1 = (const float*)d_in[3];
    const float* g1 = (const float*)d_in[4];
    const float* be1 = (const float*)d_in[5];
    const float* W2 = (const float*)d_in[6];
    const float* b2 = (const float*)d_in[7];
    const float* g2 = (const float*)d_in[8];
    const float* be2 = (const float*)d_in[9];
    const float* W3 = (const float*)d_in[10];
    const float* b3 = (const float*)d_in[11];
    float* out = (float*)d_out;
    (void)in_sizes; (void)n_in; (void)out_size; (void)d_ws; (void)ws_size;

    dim3 grid(kBatch / kRowsPerBlock);
    dim3 block(kThreads);
    walsh_encoder_kernel<<<grid, block, 0, stream>>>(
        tt, W1, b1, g1, be1, W2, b2, g2, be2, W3, b3, out);
}